// ContrastiveLoss_10488310136947
// MI455X (gfx1250) — compile-verified
//
#include <hip/hip_runtime.h>

typedef __attribute__((ext_vector_type(2))) float v2f;
typedef __attribute__((ext_vector_type(8))) float v8f;

namespace {
constexpr int kB = 8, kC = 64, kW = 256, kH = 256;
constexpr int kWH = kW * kH;
constexpr int kN = 1024, kM = 16;
constexpr float kMargin = 0.5f;
constexpr int kThreads = 256;                               // 8 wave32s
constexpr int kPtsPerBlock = 128;                           // 8 waves * 16 pts
constexpr int kPosBlocks = (kB * kN) / kPtsPerBlock;        // 64
constexpr int kNegBlocks = (kB * kM * kN) / kPtsPerBlock;   // 1024
// d_ws float layout
constexpr int kPosOff = 0;                  // [0, 64)
constexpr int kNegOff = kPosBlocks;         // [64, 1088)
constexpr int kCntOff = kNegOff + kNegBlocks; // [1088, 2112)
}

// Per-wave: 16 points, dist_i = diag_i of (A-B)(A-B)^T via f32 WMMA.
// p1/p2 are per-lane channel-0 pointers for row r = lane&15; lanes 16..31
// mirror rows 0..15 but feed K-halves {k0+2,k0+3} per the 16x4 f32 A layout.
__device__ __forceinline__ v8f wave_dist16(const float* __restrict__ p1,
                                           const float* __restrict__ p2,
                                           int lane) {
  v8f acc = {0.f, 0.f, 0.f, 0.f, 0.f, 0.f, 0.f, 0.f};
#if __has_builtin(__builtin_amdgcn_wmma_f32_16x16x4_f32)
  const int khalf = (lane >= 16) ? 2 : 0;
#pragma unroll
  for (int k0 = 0; k0 < kC; k0 += 4) {
    const int ch = k0 + khalf;
    const float m0 = p1[(size_t)ch * kWH] - p2[(size_t)ch * kWH];
    const float m1 = p1[(size_t)(ch + 1) * kWH] - p2[(size_t)(ch + 1) * kWH];
    v2f a;
    a.x = m0;
    a.y = m1;
    // D = A*A^T + C ; symmetric operand => same VGPRs are valid A and B layouts
    acc = __builtin_amdgcn_wmma_f32_16x16x4_f32(false, a, false, a,
                                                (short)0, acc, false, false);
  }
#else
  // Scalar fallback: full 64-channel distance per lane, replicated into all
  // 8 accumulator slots so diagonal extraction below stays correct.
  float d = 0.f;
#pragma unroll 8
  for (int ch = 0; ch < kC; ++ch) {
    const float m = p1[(size_t)ch * kWH] - p2[(size_t)ch * kWH];
    d += m * m;
  }
#pragma unroll
  for (int v = 0; v < 8; ++v) acc[v] = d;
#endif
  return acc;
}

// Extract diagonal of 16x16 f32 C-layout; writes 16 dists to sdist[0..15].
__device__ __forceinline__ void diag_to_lds(const v8f& acc, int lane,
                                            float* sdist16) {
  const int sel = lane & 7;  // lanes 0..7: vgpr=lane ; lanes 24..31: vgpr=lane-24
  float d = acc[0];
  d = (sel == 1) ? acc[1] : d;
  d = (sel == 2) ? acc[2] : d;
  d = (sel == 3) ? acc[3] : d;
  d = (sel == 4) ? acc[4] : d;
  d = (sel == 5) ? acc[5] : d;
  d = (sel == 6) ? acc[6] : d;
  d = (sel == 7) ? acc[7] : d;
  const bool active = (lane < 8) || (lane >= 24);
  const int i = (lane < 8) ? lane : (lane - 16);  // diag index 0..15
  if (active) sdist16[i] = d;
}

__device__ __forceinline__ float block_reduce_sum(float v, float* red) {
  const int tid = threadIdx.x;
  __syncthreads();  // protect red[] reuse across back-to-back reductions
  red[tid] = v;
  __syncthreads();
#pragma unroll
  for (int s = kThreads / 2; s > 0; s >>= 1) {
    if (tid < s) red[tid] += red[tid + s];
    __syncthreads();
  }
  return red[0];
}

// ---- Kernel 1: positive pairs. 64 blocks * 128 points. ----
__global__ __launch_bounds__(kThreads) void cl_pos_kernel(
    const float* __restrict__ out1, const float* __restrict__ out2,
    const int* __restrict__ match1, const int* __restrict__ match2,
    float* __restrict__ ws) {
  __shared__ float sdist[kPtsPerBlock];
  __shared__ float red[kThreads];

  const int lane = threadIdx.x & 31;
  const int wib = threadIdx.x >> 5;
  const int point = (blockIdx.x * 8 + wib) * 16 + (lane & 15);  // [0, B*N)
  const int bIdx = point >> 10;  // / kN

  const int x1 = match1[point * 2 + 0];
  const int y1 = match1[point * 2 + 1];
  const int x2 = match2[point * 2 + 0];
  const int y2 = match2[point * 2 + 1];
  const size_t bbase = (size_t)bIdx * kC * kWH;
  const float* p1 = out1 + bbase + (size_t)x1 * kH + y1;
  const float* p2 = out2 + bbase + (size_t)x2 * kH + y2;

  const v8f acc = wave_dist16(p1, p2, lane);
  diag_to_lds(acc, lane, &sdist[wib * 16]);
  __syncthreads();

  const float v = (threadIdx.x < kPtsPerBlock) ? sdist[threadIdx.x] : 0.f;
  const float s = block_reduce_sum(v, red);
  if (threadIdx.x == 0) ws[kPosOff + blockIdx.x] = s;
}

// ---- Kernel 2: negative pairs. 1024 blocks * 128 points. ----
__global__ __launch_bounds__(kThreads) void cl_neg_kernel(
    const float* __restrict__ out1, const float* __restrict__ out2,
    const int* __restrict__ match1, const int* __restrict__ nonmatch2,
    float* __restrict__ ws) {
  __shared__ float sdist[kPtsPerBlock];
  __shared__ float red[kThreads];

  const int lane = threadIdx.x & 31;
  const int wib = threadIdx.x >> 5;
  const int point = (blockIdx.x * 8 + wib) * 16 + (lane & 15);  // [0, B*M*N)
  const int bIdx = point >> 14;        // / (M*N)
  const int j = point & (kM * kN - 1); // index within (m,n)
  const int jn = j & (kN - 1);         // f1_rep: tile over m -> j % n

  const int m1i = (bIdx * kN + jn) * 2;
  const int x1 = match1[m1i + 0];
  const int y1 = match1[m1i + 1];
  const int x2 = nonmatch2[point * 2 + 0];
  const int y2 = nonmatch2[point * 2 + 1];
  const size_t bbase = (size_t)bIdx * kC * kWH;
  const float* p1 = out1 + bbase + (size_t)x1 * kH + y1;
  const float* p2 = out2 + bbase + (size_t)x2 * kH + y2;

  const v8f acc = wave_dist16(p1, p2, lane);
  diag_to_lds(acc, lane, &sdist[wib * 16]);
  __syncthreads();

  float hinge = 0.f, cnt = 0.f;
  if (threadIdx.x < kPtsPerBlock) {
    const float diff = kMargin - sdist[threadIdx.x];
    if (diff > 0.f) {
      hinge = diff;
      cnt = 1.f;
    }
  }
  const float hs = block_reduce_sum(hinge, red);
  const float cs = block_reduce_sum(cnt, red);
  if (threadIdx.x == 0) {
    ws[kNegOff + blockIdx.x] = hs;
    ws[kCntOff + blockIdx.x] = cs;
  }
}

// ---- Kernel 3: deterministic final reduction of partials. ----
__global__ __launch_bounds__(kThreads) void cl_finalize_kernel(
    const float* __restrict__ ws, float* __restrict__ out) {
  __shared__ float red[kThreads];
  const int tid = threadIdx.x;

  float s0 = 0.f, s1 = 0.f, s2 = 0.f;
  for (int i = tid; i < kPosBlocks; i += kThreads) s0 += ws[kPosOff + i];
  for (int i = tid; i < kNegBlocks; i += kThreads) s1 += ws[kNegOff + i];
  for (int i = tid; i < kNegBlocks; i += kThreads) s2 += ws[kCntOff + i];

  const float pos_sum = block_reduce_sum(s0, red);
  const float neg_sum = block_reduce_sum(s1, red);
  const float n_hard = block_reduce_sum(s2, red);

  if (tid == 0) {
    const float pos_loss = pos_sum / (float)(kB * kN);
    const float neg_loss = neg_sum / n_hard;  // matches reference (no guard)
    out[0] = pos_loss + neg_loss;
  }
}

extern "C" void kernel_launch(void* const* d_in, const int* in_sizes, int n_in,
                              void* d_out, int out_size, void* d_ws, size_t ws_size,
                              hipStream_t stream) {
  (void)in_sizes; (void)n_in; (void)out_size; (void)ws_size;
  const float* out1 = (const float*)d_in[0];
  const float* out2 = (const float*)d_in[1];
  const int* match1 = (const int*)d_in[2];
  const int* match2 = (const int*)d_in[3];
  const int* nonmatch2 = (const int*)d_in[4];
  float* out = (float*)d_out;
  float* ws = (float*)d_ws;

  cl_pos_kernel<<<kPosBlocks, kThreads, 0, stream>>>(out1, out2, match1,
                                                     match2, ws);
  cl_neg_kernel<<<kNegBlocks, kThreads, 0, stream>>>(out1, out2, match1,
                                                     nonmatch2, ws);
  cl_finalize_kernel<<<1, kThreads, 0, stream>>>(ws, out);
}